// CoincidenceLIFBank_81698867905128
// MI455X (gfx1250) — compile-verified
//
#include <hip/hip_runtime.h>
#include <cstdint>

// Problem constants (from reference: B=32, T=4096, D=256)
constexpr int kB  = 32;
constexpr int kT  = 4096;
constexpr int kD  = 256;          // total delay channels
constexpr int kDS = 4;            // D split across blocks -> 128 blocks total
constexpr int kDL = kD / kDS;     // 64 channels (threads) per block = 2 wave32 waves
constexpr int kTC = 32;           // t-steps per chunk (chunked sequential scan)
constexpr int kQG = kTC / 4;      // float4 groups per d-row per chunk = 8
constexpr int kSROW = kTC + 4;    // padded LDS row stride (floats) -> <=2-way bank conflict
constexpr int kNCHUNK = kT / kTC; // 128

// LDS layout (floats): ref row [0,4096) | tgt row [4096,8192) | 2 spike buffers of 64*36
constexpr size_t kShmemFloats = 2 * kT + 2 * kDL * kSROW;  // 12800 floats = 51200 B (<320KB WGP LDS)

__global__ __launch_bounds__(kDL) void CoincidenceLIFBank_kernel(
    const float* __restrict__ ref,    // [B,T]
    const float* __restrict__ tgt,    // [B,T]
    const int*   __restrict__ delays, // [D]
    const float* __restrict__ rw_raw, // [D]
    const float* __restrict__ tw_raw, // [D]
    const float* __restrict__ bt_raw, // [D]
    float* __restrict__ out)          // pooled[B*D] | spikes[B*D*T] | rw[D] | tw[D] | beta[D]
{
    extern __shared__ float smem[];
    float* ref_s = smem;               // 4096 floats
    float* tgt_s = smem + kT;          // 4096 floats
    float* spk_s = smem + 2 * kT;      // 2 * 64 * 36 floats

    const int tid = threadIdx.x;              // local channel
    const int b   = blockIdx.x;               // batch row
    const int d   = blockIdx.y * kDL + tid;   // global channel

    // LDS byte addresses (flat low 32 bits == wave-relative LDS offset)
    const uint32_t ref_lb = (uint32_t)(uintptr_t)(void*)ref_s;
    const uint32_t tgt_lb = (uint32_t)(uintptr_t)(void*)tgt_s;
    const uint32_t spk_lb = (uint32_t)(uintptr_t)(void*)spk_s;

    // ---- Stage input rows into LDS via async global->LDS DMA (ASYNCcnt) ----
    {
        const uint32_t rowoff = (uint32_t)(b * kT) * 4u;   // byte offset of row b
#pragma unroll
        for (int i = 0; i < kT * 4 / (kDL * 16); ++i) {    // 16 iters * 64 lanes * 16B = 16KB
            const uint32_t c16 = (uint32_t)(i * kDL + tid) * 16u;
            const uint32_t go  = rowoff + c16;
            const uint32_t lo  = ref_lb + c16;
            asm volatile("global_load_async_to_lds_b128 %0, %1, %2"
                         :: "v"(lo), "v"(go), "s"(ref) : "memory");
        }
#pragma unroll
        for (int i = 0; i < kT * 4 / (kDL * 16); ++i) {
            const uint32_t c16 = (uint32_t)(i * kDL + tid) * 16u;
            const uint32_t go  = rowoff + c16;
            const uint32_t lo  = tgt_lb + c16;
            asm volatile("global_load_async_to_lds_b128 %0, %1, %2"
                         :: "v"(lo), "v"(go), "s"(tgt) : "memory");
        }
    }

    // ---- Per-channel parameters (normal loads; overlap with the DMA in flight) ----
    const int   delay = delays[d];
    const float rw    = log1pf(expf(rw_raw[d]));           // softplus
    const float tw    = log1pf(expf(tw_raw[d]));
    const float beta  = 0.7f + 0.295f * (1.0f / (1.0f + expf(-bt_raw[d])));

    asm volatile("s_wait_asynccnt 0x0" ::: "memory");      // drain staging DMA
    __syncthreads();                                       // rows visible block-wide

    float m   = 0.0f;   // membrane (>= 0 by induction: inputs and weights non-negative)
    float cnt = 0.0f;   // spike count for pooled mean

    float* const out_spk = out + (size_t)kB * kD;          // spike tensor base

    // ---- Chunked sequential LIF scan with double-buffered async store-out ----
    for (int chunk = 0; chunk < kNCHUNK; ++chunk) {
        const int buf = chunk & 1;
        const int t0  = chunk * kTC;

        // Buffer reuse gate: allow only the newest 8 async stores (chunk-1) to be
        // outstanding; chunk-2 (same buffer) is then complete. Barrier extends the
        // guarantee to every wave before anyone overwrites the buffer.
        asm volatile("s_wait_asynccnt 0x8" ::: "memory");
        __syncthreads();

        float* const row = spk_s + buf * (kDL * kSROW) + tid * kSROW;
#pragma unroll
        for (int j = 0; j < kTC; j += 4) {
            float sv[4];
#pragma unroll
            for (int c = 0; c < 4; ++c) {
                const int   t   = t0 + j + c;
                const int   idx = t + delay;                          // delay in [0,128)
                const float rv  = (idx < kT) ? ref_s[idx] : 0.0f;     // shift + validity mask
                const float cur = fmaf(rw, rv, tw * tgt_s[t]);        // tw*tgt off critical path
                m = fmaf(beta, m, cur);                               // carried dep 1: fma
                const bool fire = (m >= 1.0f);                        // carried dep 2: cmp
                m = fire ? (m - 1.0f) : m;   // == max(m - spike, 0) since m >= 0  (dep 3: cndmask)
                sv[c] = fire ? 1.0f : 0.0f;  // off-chain selects
                cnt  += fire ? 1.0f : 0.0f;
            }
            float4 s4; s4.x = sv[0]; s4.y = sv[1]; s4.z = sv[2]; s4.w = sv[3];
            *reinterpret_cast<float4*>(row + j) = s4;                 // ds_store_b128
        }
        __syncthreads();   // DScnt drain + barrier: buffer fully written by all waves

        // DMA chunk LDS -> global: 8 consecutive lanes cover one d-row's 8 float4
        // groups => 128B contiguous global segments (cacheline-granular stores).
        const uint32_t bufbase = spk_lb + (uint32_t)(buf * (kDL * kSROW * 4));
#pragma unroll
        for (int i = 0; i < kQG; ++i) {                    // 8 async stores per wave per chunk
            const uint32_t c  = (uint32_t)(i * kDL + tid); // [0, 512)
            const uint32_t dd = c >> 3;                    // local d row [0,64)
            const uint32_t q  = c & 7u;                    // float4 group within chunk
            const uint32_t dg = (uint32_t)(blockIdx.y * kDL) + dd;    // global d
            const uint32_t go = (((uint32_t)(b * kD) + dg) * (uint32_t)kT
                                 + (uint32_t)t0 + q * 4u) * 4u;
            const uint32_t lo = bufbase + (dd * kSROW + q * 4u) * 4u;
            asm volatile("global_store_async_from_lds_b128 %0, %1, %2"
                         :: "v"(go), "v"(lo), "s"(out_spk) : "memory");
        }
    }
    asm volatile("s_wait_asynccnt 0x0" ::: "memory");      // drain all DMA stores

    // ---- Pooled mean over T ----
    out[b * kD + d] = cnt * (1.0f / (float)kT);

    // ---- Diagnostics (written once: the b==0 column of blocks covers all d) ----
    if (b == 0) {
        float* const diag = out + (size_t)kB * kD + (size_t)kB * kD * kT;
        diag[d]            = rw;
        diag[kD + d]       = tw;
        diag[2 * kD + d]   = beta;
    }
}

extern "C" void kernel_launch(void* const* d_in, const int* in_sizes, int n_in,
                              void* d_out, int out_size, void* d_ws, size_t ws_size,
                              hipStream_t stream) {
    (void)in_sizes; (void)n_in; (void)d_ws; (void)ws_size; (void)out_size;
    const float* ref    = (const float*)d_in[0];
    const float* tgt    = (const float*)d_in[1];
    const int*   delays = (const int*)  d_in[2];
    const float* rww    = (const float*)d_in[3];
    const float* tww    = (const float*)d_in[4];
    const float* btw    = (const float*)d_in[5];
    float* out = (float*)d_out;

    const size_t shmem = kShmemFloats * sizeof(float);     // 51200 B per block
    dim3 grid(kB, kDS);                                    // 32 x 4 = 128 blocks -> many WGPs
    CoincidenceLIFBank_kernel<<<grid, kDL, shmem, stream>>>(ref, tgt, delays, rww, tww, btw, out);
}